// GCN_23055384445692
// MI455X (gfx1250) — compile-verified
//
#include <hip/hip_runtime.h>

#define HID 256

typedef __attribute__((ext_vector_type(2))) float v2f;
typedef __attribute__((ext_vector_type(8))) float v8f;

// ---------------------------------------------------------------------------
// Utility: zero a float buffer (float4 granularity; callers pass n%4==0)
// ---------------------------------------------------------------------------
__global__ void zero_f32_kernel(float4* __restrict__ p, int n4) {
    int i = blockIdx.x * blockDim.x + threadIdx.x;
    if (i < n4) p[i] = make_float4(0.f, 0.f, 0.f, 0.f);
}

// ---------------------------------------------------------------------------
// Degrees: atomic float add per edge endpoint
// ---------------------------------------------------------------------------
__global__ void degree_kernel(const int* __restrict__ src, const int* __restrict__ dst,
                              float* __restrict__ degO, float* __restrict__ degI, int nE) {
    int e = blockIdx.x * blockDim.x + threadIdx.x;
    if (e < nE) {
        atomicAdd(&degO[src[e]], 1.0f);
        atomicAdd(&degI[dst[e]], 1.0f);
    }
}

// In-place: deg -> (deg>0 ? deg : 1)^-0.5
__global__ void norm_kernel(float* __restrict__ degO, float* __restrict__ degI, int n) {
    int i = blockIdx.x * blockDim.x + threadIdx.x;
    if (i < n) {
        float a = degO[i];
        degO[i] = 1.0f / sqrtf(a > 0.f ? a : 1.0f);
        float b = degI[i];
        degI[i] = 1.0f / sqrtf(b > 0.f ? b : 1.0f);
    }
}

// ---------------------------------------------------------------------------
// SpMM scatter: agg[dst[e], :] += h[src[e], :]   (h is pre-scaled by norm_out)
// 64 threads per edge, one float4 per thread; a wave covers 128 contiguous
// features of a single edge -> fully coalesced gather, 4 f32 atomics/lane.
// ---------------------------------------------------------------------------
__global__ void scatter_kernel(const float* __restrict__ h, const int* __restrict__ src,
                               const int* __restrict__ dst, float* __restrict__ agg,
                               unsigned nE) {
    unsigned gid = blockIdx.x * blockDim.x + threadIdx.x;
    unsigned e = gid >> 6;
    if (e >= nE) return;
    int c = (int)(gid & 63u) << 2;
    int s = src[e], d = dst[e];
    const float4 v = *(const float4*)(h + (size_t)s * HID + c);
    float* p = agg + (size_t)d * HID + c;
    atomicAdd(p + 0, v.x);
    atomicAdd(p + 1, v.y);
    atomicAdd(p + 2, v.z);
    atomicAdd(p + 3, v.w);
}

// ---------------------------------------------------------------------------
// Layer-0 epilogue fused with layer-1 gather prescale:
//   hout[i,:] = relu(agg[i,:]*norm_in[i] + bias) * norm_out[i]
// (norm_out >= 0, so scaling after relu == relu of scaled pre-activation path)
// ---------------------------------------------------------------------------
__global__ void epilogue_kernel(const float* __restrict__ agg, const float* __restrict__ normI,
                                const float* __restrict__ normO, const float* __restrict__ bias,
                                float* __restrict__ hout, unsigned n) {
    unsigned gid = blockIdx.x * blockDim.x + threadIdx.x;
    unsigned row = gid >> 6;
    if (row >= n) return;
    int c = (int)(gid & 63u) << 2;
    float ni = normI[row];
    float no = normO[row];
    float4 v = *(const float4*)(agg + (size_t)row * HID + c);
    float4 b = *(const float4*)(bias + c);
    float4 r;
    r.x = fmaxf(fmaf(v.x, ni, b.x), 0.f) * no;
    r.y = fmaxf(fmaf(v.y, ni, b.y), 0.f) * no;
    r.z = fmaxf(fmaf(v.z, ni, b.z), 0.f) * no;
    r.w = fmaxf(fmaf(v.w, ni, b.w), 0.f) * no;
    *(float4*)(hout + (size_t)row * HID + c) = r;
}

// ---------------------------------------------------------------------------
// FP32 WMMA GEMM:  out[M,256] = f(A[M,K] (row-scaled) @ W[K,256] + bias)
//   - one wave computes a 16x64 strip: A-fragment reused across 4 WMMA tiles
//   - V_WMMA_F32_16X16X4_F32, K stepped by 4
//   - ascale: optional per-row scale on A (norm_in for layer 1)
//   - oscale: optional per-row scale on output (norm_out for projections)
//   - EXEC stays all-ones around WMMA: loads clamped, stores guarded
// ---------------------------------------------------------------------------
__device__ __forceinline__ void store_tile(v8f acc, float* __restrict__ out, int M,
                                           int mt, int half, int col, float bv, int relu,
                                           const float* __restrict__ oscale) {
#pragma unroll
    for (int i = 0; i < 8; ++i) {
        int row = mt * 16 + half * 8 + i;   // C/D layout: VGPR i -> M=i (+8 for lanes 16-31)
        if (row < M) {
            float v = acc[i] + bv;
            if (relu) v = fmaxf(v, 0.f);
            if (oscale) v *= oscale[row];
            out[(size_t)row * HID + col] = v;
        }
    }
}

__global__ void wmma_gemm_kernel(const float* __restrict__ A, const float* __restrict__ W,
                                 const float* __restrict__ bias,
                                 const float* __restrict__ ascale,
                                 const float* __restrict__ oscale,
                                 float* __restrict__ out, int M, int K, int relu) {
    const int wid  = threadIdx.x >> 5;
    const int lane = threadIdx.x & 31;
    const int numMTiles = (M + 15) >> 4;
    const int task = blockIdx.x * 8 + wid;           // 8 waves per block
    if (task >= numMTiles * 4) return;               // wave-uniform exit
    const int mt    = task >> 2;
    const int ncol0 = (task & 3) * 64;               // this wave's 4 N-tiles
    const int half  = lane >> 4;                     // 0: K=kb,kb+1  1: K=kb+2,kb+3
    const int l     = lane & 15;

    int arow = mt * 16 + l;
    if (arow >= M) arow = M - 1;                     // clamp loads; stores guarded
    const float as = ascale ? ascale[arow] : 1.0f;

    const float* Ap = A + (size_t)arow * K + 2 * half;
    const float* Wp = W + (size_t)(2 * half) * HID + ncol0 + l;

    v8f acc0 = {}, acc1 = {}, acc2 = {}, acc3 = {};
    for (int kb = 0; kb < K; kb += 4) {
        v2f a, b0, b1, b2, b3;
        a.x = Ap[kb] * as;                           // A 16x4 fp32 fragment
        a.y = Ap[kb + 1] * as;
        const float* w0 = Wp + (size_t)kb * HID;
        const float* w1 = w0 + HID;
        b0.x = w0[0];  b0.y = w1[0];                 // B 4x16 fragments, 4 N-tiles
        b1.x = w0[16]; b1.y = w1[16];
        b2.x = w0[32]; b2.y = w1[32];
        b3.x = w0[48]; b3.y = w1[48];
        acc0 = __builtin_amdgcn_wmma_f32_16x16x4_f32(false, a, false, b0, (short)0, acc0, false, false);
        acc1 = __builtin_amdgcn_wmma_f32_16x16x4_f32(false, a, false, b1, (short)0, acc1, false, false);
        acc2 = __builtin_amdgcn_wmma_f32_16x16x4_f32(false, a, false, b2, (short)0, acc2, false, false);
        acc3 = __builtin_amdgcn_wmma_f32_16x16x4_f32(false, a, false, b3, (short)0, acc3, false, false);
    }

    const int col = ncol0 + l;
    store_tile(acc0, out, M, mt, half, col +  0, bias[col +  0], relu, oscale);
    store_tile(acc1, out, M, mt, half, col + 16, bias[col + 16], relu, oscale);
    store_tile(acc2, out, M, mt, half, col + 32, bias[col + 32], relu, oscale);
    store_tile(acc3, out, M, mt, half, col + 48, bias[col + 48], relu, oscale);
}

// ---------------------------------------------------------------------------
// Orchestration
// ---------------------------------------------------------------------------
static inline void launch_gemm(const float* A, const float* W, const float* b, int M, int K,
                               const float* ascale, const float* oscale, float* outp,
                               int relu, hipStream_t stream) {
    int tasks = ((M + 15) / 16) * 4;
    int blocks = (tasks + 7) / 8;
    wmma_gemm_kernel<<<blocks, 256, 0, stream>>>(A, W, b, ascale, oscale, outp, M, K, relu);
}

extern "C" void kernel_launch(void* const* d_in, const int* in_sizes, int n_in,
                              void* d_out, int out_size, void* d_ws, size_t ws_size,
                              hipStream_t stream) {
    const float* x0    = (const float*)d_in[0];
    const float* x1    = (const float*)d_in[1];
    const float* x2    = (const float*)d_in[2];
    const float* W_fc0 = (const float*)d_in[3];
    const float* b_fc0 = (const float*)d_in[4];
    const float* W_fc1 = (const float*)d_in[5];
    const float* b_fc1 = (const float*)d_in[6];
    const float* W_fc2 = (const float*)d_in[7];
    const float* b_fc2 = (const float*)d_in[8];
    const float* bias_l0 = (const float*)d_in[9];
    const float* W_l1  = (const float*)d_in[10];
    const float* b_l1  = (const float*)d_in[11];
    const int*   src   = (const int*)d_in[12];
    const int*   dst   = (const int*)d_in[13];
    float* out = (float*)d_out;

    const int K0 = in_sizes[3] / HID, M0 = in_sizes[0] / K0;
    const int K1 = in_sizes[5] / HID, M1 = in_sizes[1] / K1;
    const int K2 = in_sizes[7] / HID, M2 = in_sizes[2] / K2;
    const int n  = M0 + M1 + M2;          // 50000
    const int nE = in_sizes[12];          // 1600000

    // Workspace: agg[n*HID] | degO/normO[n] | degI/normI[n]
    float* agg  = (float*)d_ws;
    float* degO = agg + (size_t)n * HID;
    float* degI = degO + n;
    // h buffer lives in d_out (written by projections, consumed by scatter,
    // rewritten by epilogue, and finally overwritten by the last GEMM).
    float* hbuf = out;

    // 1) degrees -> norms
    {
        int n4 = (2 * n + 3) / 4;         // degO+degI contiguous
        zero_f32_kernel<<<(n4 + 255) / 256, 256, 0, stream>>>((float4*)degO, n4);
    }
    degree_kernel<<<(nE + 255) / 256, 256, 0, stream>>>(src, dst, degO, degI, nE);
    norm_kernel<<<(n + 255) / 256, 256, 0, stream>>>(degO, degI, n);

    // 2) projections: hbuf[row] = (x @ W + b) * norm_out[row]   (WMMA fp32)
    launch_gemm(x0, W_fc0, b_fc0, M0, K0, nullptr, degO,            hbuf,                        0, stream);
    launch_gemm(x1, W_fc1, b_fc1, M1, K1, nullptr, degO + M0,       hbuf + (size_t)M0 * HID,      0, stream);
    launch_gemm(x2, W_fc2, b_fc2, M2, K2, nullptr, degO + M0 + M1,  hbuf + (size_t)(M0 + M1) * HID, 0, stream);

    // 3) layer 0 SpMM
    {
        int n4 = n * HID / 4;
        zero_f32_kernel<<<(n4 + 255) / 256, 256, 0, stream>>>((float4*)agg, n4);
    }
    {
        unsigned long long th = (unsigned long long)nE * 64ull;
        unsigned blocks = (unsigned)((th + 255ull) / 256ull);
        scatter_kernel<<<blocks, 256, 0, stream>>>(hbuf, src, dst, agg, (unsigned)nE);
    }
    // hbuf[i] = relu(agg[i]*norm_in + bias_l0) * norm_out[i]  (prescaled for layer-1 gather)
    {
        unsigned long long th = (unsigned long long)n * 64ull;
        unsigned blocks = (unsigned)((th + 255ull) / 256ull);
        epilogue_kernel<<<blocks, 256, 0, stream>>>(agg, degI, degO, bias_l0, hbuf, (unsigned)n);
    }

    // 4) layer 1 SpMM
    {
        int n4 = n * HID / 4;
        zero_f32_kernel<<<(n4 + 255) / 256, 256, 0, stream>>>((float4*)agg, n4);
    }
    {
        unsigned long long th = (unsigned long long)nE * 64ull;
        unsigned blocks = (unsigned)((th + 255ull) / 256ull);
        scatter_kernel<<<blocks, 256, 0, stream>>>(hbuf, src, dst, agg, (unsigned)nE);
    }

    // 5) final GEMM: out = relu((agg * norm_in[:,None]) @ W_l1 + b_l1)
    //    norm_in fused into A-fragment load; bias+relu fused into store.
    launch_gemm(agg, W_l1, b_l1, n, HID, degI, nullptr, out, 1, stream);
}